// AutoEncoder_32504312496394
// MI455X (gfx1250) — compile-verified
//
#include <hip/hip_runtime.h>
#include <hip/hip_bf16.h>

typedef float v2f __attribute__((ext_vector_type(2)));
typedef float v8f __attribute__((ext_vector_type(8)));

#define NN 50000
#define NE 800000
#define EPS 1e-5f

__device__ __forceinline__ void atomAddF(float* p, float v) {
  __hip_atomic_fetch_add(p, v, __ATOMIC_RELAXED, __HIP_MEMORY_SCOPE_AGENT);
}

// ---------------- degree / norm precompute ----------------
__global__ void k_init_deg(float* __restrict__ deg) {
  int i = blockIdx.x * 256 + threadIdx.x;
  if (i < NN) deg[i] = 1.0f;  // self-loop contributes 1 to degree
}

__global__ void k_count_deg(const int* __restrict__ dst, float* __restrict__ deg) {
  int e = blockIdx.x * 256 + threadIdx.x;
  if (e < NE) atomAddF(&deg[dst[e]], 1.0f);
}

__global__ void k_dis(float* __restrict__ deg) {
  int i = blockIdx.x * 256 + threadIdx.x;
  if (i < NN) {
    float v = deg[i];
    deg[i] = (v > 0.0f) ? rsqrtf(v) : 0.0f;  // deg -> dis in place
  }
}

__global__ void k_norm(const int* __restrict__ src, const int* __restrict__ dst,
                       const float* __restrict__ dis, float* __restrict__ norm) {
  int e = blockIdx.x * 256 + threadIdx.x;
  if (e < NE) norm[e] = dis[src[e]] * dis[dst[e]];
}

// ---------------- zero-pad W[K,N] -> Wp[Kp,Np] ----------------
template <int K, int N>
__global__ void k_padW(const float* __restrict__ W, float* __restrict__ Wp) {
  constexpr int Kp = (K + 3) & ~3;
  constexpr int Np = (N + 15) & ~15;
  int id = blockIdx.x * 256 + threadIdx.x;
  if (id < Kp * Np) {
    int k = id / Np;
    int n = id - k * Np;
    Wp[id] = (k < K && n < N) ? W[k * N + n] : 0.0f;
  }
}

// ---------------- fp32 WMMA GEMM: H[M,N] = X[M,K] @ Wp[Kp,Np] ----------------
// One wave per 16-row block; computes ALL N-tiles so each A fragment is loaded
// once and reused. Fully unrolled, branch-free inner loop (zero-padded operands).
template <int K, int N>
__global__ __launch_bounds__(32) void k_gemm(const float* __restrict__ X, int lda,
                                             const float* __restrict__ Wp,
                                             float* __restrict__ H, int ldh) {
  constexpr int Kp = (K + 3) & ~3;
  constexpr int Np = (N + 15) & ~15;
  constexpr int NT = Np / 16;
  const int lane = threadIdx.x;      // 0..31
  const int half = lane >> 4;        // K-half select (ISA 16x4 f32 A layout)
  const int l16  = lane & 15;
  const int rowA = blockIdx.x * 16 + l16;  // always < 50000 (3125*16 tiles)

  v8f acc[NT];
#pragma unroll
  for (int t = 0; t < NT; ++t) {
    v8f z = {0.f, 0.f, 0.f, 0.f, 0.f, 0.f, 0.f, 0.f};
    acc[t] = z;
  }

  const float* xrow = X + (size_t)rowA * lda;
#pragma unroll
  for (int kk = 0; kk < Kp; kk += 4) {
    const int ka = kk + 2 * half;                    // even -> 8B aligned
    v2f a = *(const v2f*)(xrow + ka);
#pragma unroll
    for (int t = 0; t < NT; ++t) {
      v2f b;
      b[0] = Wp[ka * Np + t * 16 + l16];
      b[1] = Wp[(ka + 1) * Np + t * 16 + l16];
      acc[t] = __builtin_amdgcn_wmma_f32_16x16x4_f32(false, a, false, b,
                                                     (short)0, acc[t], false, false);
    }
  }

#pragma unroll
  for (int t = 0; t < NT; ++t) {
    int col = t * 16 + l16;
    if (col < N) {
#pragma unroll
      for (int v = 0; v < 8; ++v) {
        // C/D layout: lane covers column l16, VGPR v holds row v + 8*half
        H[(size_t)(blockIdx.x * 16 + v + half * 8) * ldh + col] = acc[t][v];
      }
    }
  }
}

// ---------------- out = h * dis^2 (self-loop) + bias (float2) ----------------
template <int F>
__global__ void k_selfbias(const float* __restrict__ h, int ldh,
                           const float* __restrict__ dis, const float* __restrict__ b,
                           float* __restrict__ out, int ldo) {
  constexpr int FH = F / 2;
  int id = blockIdx.x * 256 + threadIdx.x;
  if (id < NN * FH) {
    int i = id / FH;
    int f2 = (id - i * FH) * 2;
    float dd = dis[i];
    dd *= dd;
    float2 hv = *(const float2*)(h + i * ldh + f2);
    float2 o;
    o.x = hv.x * dd + b[f2];
    o.y = hv.y * dd + b[f2 + 1];
    *(float2*)(out + i * ldo + f2) = o;
  }
}

// ---------------- edge scatter: out[dst] += h[src] * norm (float2 gather) -----
template <int F>
__global__ void k_agg(const float* __restrict__ h, int ldh,
                      const int* __restrict__ src, const int* __restrict__ dst,
                      const float* __restrict__ norm,
                      float* __restrict__ out, int ldo) {
  constexpr int FH = F / 2;
  int id = blockIdx.x * 256 + threadIdx.x;
  if (id < NE * FH) {
    int e = id / FH;                // consecutive lanes -> consecutive features
    int f2 = (id - e * FH) * 2;
    int s = src[e];
    int d = dst[e];
    float w = norm[e];
    float2 hv = *(const float2*)(h + s * ldh + f2);
    atomAddF(out + d * ldo + f2,     hv.x * w);
    atomAddF(out + d * ldo + f2 + 1, hv.y * w);
  }
}

// ---------------- zero pad columns [c0, c0+ncols) so next GEMM is guard-free --
__global__ void k_zeropad(float* __restrict__ buf, int ld, int c0, int ncols) {
  int id = blockIdx.x * 256 + threadIdx.x;
  if (id < NN * ncols) {
    int i = id / ncols;
    int c = id - i * ncols;
    buf[i * ld + c0 + c] = 0.0f;
  }
}

// ---------------- zero the 192-float stats region ----------------
__global__ void k_zero(float* __restrict__ p) {
  if (threadIdx.x < 192) p[threadIdx.x] = 0.0f;
}

// -------- fused ReLU + BN stats (LDS ds_add_f32, then few global atomics) -----
__global__ void k_relustats(float* __restrict__ h, int ld, int F,
                            float* __restrict__ gsum, float* __restrict__ gsq) {
  __shared__ float ssum[96];
  __shared__ float ssq[96];
  int tid = threadIdx.x;
  for (int f = tid; f < 96; f += 256) { ssum[f] = 0.0f; ssq[f] = 0.0f; }
  __syncthreads();

  int r0 = blockIdx.x * 64;
  int rend = r0 + 64;
  if (rend > NN) rend = NN;
  int nwork = (rend - r0) * F;
  for (int idx = tid; idx < nwork; idx += 256) {
    int q = idx / F;
    int f = idx - q * F;
    int i = r0 + q;
    float v = h[i * ld + f];
    v = (v > 0.0f) ? v : 0.0f;  // ReLU in place
    h[i * ld + f] = v;
    atomicAdd(&ssum[f], v);
    atomicAdd(&ssq[f], v * v);
  }
  __syncthreads();
  for (int f = tid; f < F; f += 256) {
    atomAddF(&gsum[f], ssum[f]);
    atomAddF(&gsq[f],  ssq[f]);
  }
}

// ---------------- BN finalize: per-column scale/shift ----------------
__global__ void k_bnfinal(const float* __restrict__ gsum, const float* __restrict__ gsq,
                          const float* __restrict__ g, const float* __restrict__ be,
                          float* __restrict__ ss, int F) {
  int f = threadIdx.x;
  if (f < F) {
    float inv_n = 1.0f / (float)NN;
    float mu = gsum[f] * inv_n;
    float var = gsq[f] * inv_n - mu * mu;
    float sc = g[f] * rsqrtf(var + EPS);
    ss[f]      = sc;
    ss[96 + f] = be[f] - mu * sc;
  }
}

// ---------------- BN apply (float2) ----------------
template <int F>
__global__ void k_bnapply(float* __restrict__ h, int ld, const float* __restrict__ ss) {
  constexpr int FH = F / 2;
  int id = blockIdx.x * 256 + threadIdx.x;
  if (id < NN * FH) {
    int i = id / FH;
    int f2 = (id - i * FH) * 2;
    float2 v = *(float2*)(h + i * ld + f2);
    v.x = v.x * ss[f2]     + ss[96 + f2];
    v.y = v.y * ss[f2 + 1] + ss[96 + f2 + 1];
    *(float2*)(h + i * ld + f2) = v;
  }
}

// ---------------- per-layer driver ----------------
template <int K, int N>
static void run_layer(const float* in, int lda, const float* Wt, const float* bias,
                      float* Wpad, float* tbuf, float* obuf, int ldo,
                      const int* src, const int* dst,
                      const float* dis, const float* norm, hipStream_t stream) {
  constexpr int Kp = (K + 3) & ~3;
  constexpr int Np = (N + 15) & ~15;
  k_padW<K, N><<<(Kp * Np + 255) / 256, 256, 0, stream>>>(Wt, Wpad);
  k_gemm<K, N><<<NN / 16, 32, 0, stream>>>(in, lda, Wpad, tbuf, 96);
  k_selfbias<N><<<(NN * (N / 2) + 255) / 256, 256, 0, stream>>>(tbuf, 96, dis, bias, obuf, ldo);
  k_agg<N><<<(NE * (N / 2) + 255) / 256, 256, 0, stream>>>(tbuf, 96, src, dst, norm, obuf, ldo);
}

extern "C" void kernel_launch(void* const* d_in, const int* in_sizes, int n_in,
                              void* d_out, int out_size, void* d_ws, size_t ws_size,
                              hipStream_t stream) {
  const float* x  = (const float*)d_in[0];
  const int*   ei = (const int*)d_in[1];  // [2, NE] int
  const int* src = ei;
  const int* dst = ei + NE;

  // workspace carve-out (256B aligned blocks)
  char* w = (char*)d_ws;
  auto carve = [&](size_t bytes) {
    void* p = (void*)w;
    w += (bytes + 255) & ~(size_t)255;
    return p;
  };
  float* bufA  = (float*)carve((size_t)NN * 96 * 4);  // GEMM output (transformed feats)
  float* bufB  = (float*)carve((size_t)NN * 96 * 4);  // aggregated feats / layer io
  float* dis   = (float*)carve((size_t)NN * 4);       // deg -> D^-1/2
  float* norm  = (float*)carve((size_t)NE * 4);       // per-edge sym norm
  float* Wpad  = (float*)carve(96 * 96 * 4);          // zero-padded weights
  float* stats = (float*)carve(192 * 4);              // gsum[96] | gsq[96]
  float* ss    = (float*)carve(192 * 4);              // scale[96] | shift[96]

  // ---- per-call graph normalization ----
  k_init_deg<<<(NN + 255) / 256, 256, 0, stream>>>(dis);
  k_count_deg<<<(NE + 255) / 256, 256, 0, stream>>>(dst, dis);
  k_dis<<<(NN + 255) / 256, 256, 0, stream>>>(dis);
  k_norm<<<(NE + 255) / 256, 256, 0, stream>>>(src, dst, dis, norm);

  const float* in = x;
  int lda = 88;
  for (int l = 0; l < 6; ++l) {
    const float* Wt = (const float*)d_in[2 + 2 * l];
    const float* bs = (const float*)d_in[3 + 2 * l];
    float* obuf = (l == 5) ? (float*)d_out : bufB;
    const int ldo = (l == 5) ? 88 : 96;
    int N;

    switch (l) {
      case 0: N = 70; run_layer<88, 70>(in, lda, Wt, bs, Wpad, bufA, obuf, ldo, src, dst, dis, norm, stream); break;
      case 1: N = 60; run_layer<70, 60>(in, lda, Wt, bs, Wpad, bufA, obuf, ldo, src, dst, dis, norm, stream); break;
      case 2: N = 50; run_layer<60, 50>(in, lda, Wt, bs, Wpad, bufA, obuf, ldo, src, dst, dis, norm, stream); break;
      case 3: N = 60; run_layer<50, 60>(in, lda, Wt, bs, Wpad, bufA, obuf, ldo, src, dst, dis, norm, stream); break;
      case 4: N = 70; run_layer<60, 70>(in, lda, Wt, bs, Wpad, bufA, obuf, ldo, src, dst, dis, norm, stream); break;
      default: N = 88; run_layer<70, 88>(in, lda, Wt, bs, Wpad, bufA, obuf, ldo, src, dst, dis, norm, stream); break;
    }

    // keep padding columns of the feature buffer zero so the NEXT layer's
    // A-matrix loads (Kp = ceil(N/4)*4 columns) need no guards
    if (l < 5) {
      int Kp_next = (N + 3) & ~3;
      if (Kp_next > N) {
        int ncols = Kp_next - N;
        k_zeropad<<<(NN * ncols + 255) / 256, 256, 0, stream>>>(obuf, ldo, N, ncols);
      }
    }

    // layers 0,1,3,4: ReLU + BatchNorm
    if (l == 0 || l == 1 || l == 3 || l == 4) {
      int bi = (l <= 1) ? l : l - 1;
      const float* g  = (const float*)d_in[14 + 2 * bi];
      const float* be = (const float*)d_in[15 + 2 * bi];
      k_zero<<<1, 256, 0, stream>>>(stats);
      k_relustats<<<(NN + 63) / 64, 256, 0, stream>>>(obuf, ldo, N, stats, stats + 96);
      k_bnfinal<<<1, 128, 0, stream>>>(stats, stats + 96, g, be, ss, N);
      if (N == 70) k_bnapply<70><<<(NN * 35 + 255) / 256, 256, 0, stream>>>(obuf, ldo, ss);
      else         k_bnapply<60><<<(NN * 30 + 255) / 256, 256, 0, stream>>>(obuf, ldo, ss);
    }

    in = obuf;
    lda = ldo;
  }
}